// EdgeSparsemax_45457933861184
// MI455X (gfx1250) — compile-verified
//
#include <hip/hip_runtime.h>
#include <hip/hip_bf16.h>

typedef float v2f __attribute__((ext_vector_type(2)));
typedef float v8f __attribute__((ext_vector_type(8)));

#define CAP 1024          // max per-node segment handled by sort path (deg ~Poisson(64))
#define NEG_SENT (-3.402823466e38f)

// ---- ordered-uint encoding of float for atomicMax (exact per-node max) ----
__device__ __forceinline__ unsigned f2key(float f) {
    unsigned b = __float_as_uint(f);
    return (b & 0x80000000u) ? ~b : (b | 0x80000000u);
}
__device__ __forceinline__ float key2f(unsigned k) {
    unsigned b = (k & 0x80000000u) ? (k ^ 0x80000000u) : ~k;
    return __uint_as_float(b);
}

// ---- K0: zero per-node state (N read from device scalar; capture-safe) ----
__global__ void k_init(unsigned* __restrict__ smax_key, int* __restrict__ deg,
                       int* __restrict__ cnt, const int* __restrict__ dnum) {
    int N = *dnum;
    for (int i = blockIdx.x * blockDim.x + threadIdx.x; i < N;
         i += gridDim.x * blockDim.x) {
        smax_key[i] = 0u;   // below every encoded real float
        deg[i] = 0;
        cnt[i] = 0;
    }
}

// ---- K1: per-node max + degree via atomics ----
__global__ void k_max_deg(const float* __restrict__ logits, const int* __restrict__ dst,
                          unsigned* __restrict__ smax_key, int* __restrict__ deg, int E) {
    int e = blockIdx.x * blockDim.x + threadIdx.x;
    if (e >= E) return;
    __builtin_prefetch(logits + e + 8192, 0, 3);   // global_prefetch_b8
    __builtin_prefetch(dst + e + 8192, 0, 3);
    float s = logits[e];
    int d = dst[e];
    atomicMax(&smax_key[d], f2key(s));
    atomicAdd(&deg[d], 1);
}

// ---- K2: exclusive scan of degrees (single block, deterministic) ----
__global__ void __launch_bounds__(1024)
k_scan(const int* __restrict__ deg, int* __restrict__ cum, const int* __restrict__ dnum) {
    __shared__ int lds[1024];
    __shared__ int base_s;
    int N = *dnum;
    int tid = threadIdx.x;
    if (tid == 0) base_s = 0;
    __syncthreads();
    for (int start = 0; start < N; start += 1024) {
        int i = start + tid;
        int v = (i < N) ? deg[i] : 0;
        lds[tid] = v;
        __syncthreads();
        #pragma unroll
        for (int off = 1; off < 1024; off <<= 1) {
            int add = (tid >= off) ? lds[tid - off] : 0;
            __syncthreads();
            lds[tid] += add;
            __syncthreads();
        }
        if (i < N) cum[i] = base_s + lds[tid] - v;   // exclusive
        __syncthreads();
        if (tid == 0) base_s += lds[1023];
        __syncthreads();
    }
}

// ---- K3: counting-scatter of shifted scores into contiguous segments ----
__global__ void k_scatter(const float* __restrict__ logits, const int* __restrict__ dst,
                          const unsigned* __restrict__ smax_key, const int* __restrict__ cum,
                          int* __restrict__ cnt, float* __restrict__ gathered, int E) {
    int e = blockIdx.x * blockDim.x + threadIdx.x;
    if (e >= E) return;
    __builtin_prefetch(logits + e + 8192, 0, 3);
    float s = logits[e];
    int d = dst[e];
    float m = key2f(smax_key[d]);
    int pos = cum[d] + atomicAdd(&cnt[d], 1);
    gathered[pos] = s - m;
}

// ---- K4: per-node (one wave32 each): bitonic sort + WMMA-f32 prefix scan + tau ----
__global__ void __launch_bounds__(32)
k_tau(const float* __restrict__ gathered, const int* __restrict__ cum,
      const int* __restrict__ deg, float* __restrict__ tau, const int* __restrict__ dnum) {
    __shared__ float s_lds[CAP];
    __shared__ float cw_lds[CAP];
    __shared__ float aux[16];
    int N = *dnum;
    int lane = threadIdx.x;
    int half = (lane >> 4) & 1;    // 0: lanes 0-15, 1: lanes 16-31
    int lcol = lane & 15;

    for (int node = blockIdx.x; node < N; node += gridDim.x) {
        int dg = deg[node];                 // uniform across the wave
        if (dg == 0) { if (lane == 0) tau[node] = 0.0f; continue; }
        int base = cum[node];

        if (dg <= CAP) {
            // power-of-two segment length, minimum one 256-wide WMMA chunk
            int P = 256;
            while (P < dg) P <<= 1;

            for (int i = lane; i < P; i += 32)
                s_lds[i] = (i < dg) ? gathered[base + i] : NEG_SENT;
            __syncthreads();

            // bitonic sort, descending
            for (int k = 2; k <= P; k <<= 1) {
                for (int j = k >> 1; j > 0; j >>= 1) {
                    for (int i = lane; i < P; i += 32) {
                        int ij = i ^ j;
                        if (ij > i) {
                            float x = s_lds[i], y = s_lds[ij];
                            bool desc = ((i & k) == 0);
                            if (desc ? (x < y) : (x > y)) { s_lds[i] = y; s_lds[ij] = x; }
                        }
                    }
                    __syncthreads();
                }
            }
            // neutralize pads so prefix sums stay finite
            for (int i = dg + lane; i < P; i += 32) s_lds[i] = 0.0f;
            __syncthreads();

            // inclusive prefix sum via f32 WMMA: D = L(16x16 lower-tri) x S(16x16),
            // chained as 4x v_wmma_f32_16x16x4_f32 (exact f32 accumulate).
            float carry = 0.0f;
            for (int c = 0; c < P; c += 256) {
                v8f acc = {};
                #pragma unroll
                for (int t = 0; t < 4; ++t) {
                    int kb = t * 4 + half * 2;      // A/B VGPR layout: K split per lane-half
                    v2f a, b;
                    a.x = (kb     <= lcol) ? 1.0f : 0.0f;   // L[row][kb]
                    a.y = (kb + 1 <= lcol) ? 1.0f : 0.0f;   // L[row][kb+1]
                    b.x = s_lds[c + lcol * 16 + kb];        // S[kb][col]
                    b.y = s_lds[c + lcol * 16 + kb + 1];
                    acc = __builtin_amdgcn_wmma_f32_16x16x4_f32(
                        false, a, false, b, (short)0, acc, false, false);
                }
                // column sums live in row 15 = acc[7] of lanes 16-31
                if (half) aux[lcol] = acc[7];
                __syncthreads();
                float colpref = 0.0f;
                for (int m = 0; m < lcol; ++m) colpref += aux[m];
                float total = colpref;
                for (int m = lcol; m < 16; ++m) total += aux[m];
                float off = carry + colpref;
                #pragma unroll
                for (int r = 0; r < 8; ++r) {
                    int row = r + half * 8;          // D layout: VGPR r -> rows r / r+8
                    cw_lds[c + lcol * 16 + row] = acc[r] + off;
                }
                carry += total;
                __syncthreads();
            }

            // sparsemax support: count(rank * s_sorted > cw - 1), clamp >= 1
            int cnt_local = 0;
            for (int r = lane; r < dg; r += 32) {
                float sv = s_lds[r];
                cnt_local += ((float)(r + 1) * sv > cw_lds[r] - 1.0f) ? 1 : 0;
            }
            #pragma unroll
            for (int o = 16; o > 0; o >>= 1)
                cnt_local += __shfl_down(cnt_local, o, 32);
            if (lane == 0) {
                int supp = cnt_local < 1 ? 1 : cnt_local;
                tau[node] = (cw_lds[supp - 1] - 1.0f) / (float)supp;
            }
            __syncthreads();
        } else {
            // pathological degree: wave bisection on sum(relu(s - tau)) = 1, tau in [-1,0]
            float lo = -1.0f, hi = 0.0f;
            for (int it = 0; it < 40; ++it) {
                float mid = 0.5f * (lo + hi);
                float part = 0.0f;
                for (int i = lane; i < dg; i += 32) {
                    float v = gathered[base + i] - mid;
                    part += (v > 0.0f) ? v : 0.0f;
                }
                #pragma unroll
                for (int o = 16; o > 0; o >>= 1)
                    part += __shfl_down(part, o, 32);
                part = __shfl(part, 0, 32);
                if (part > 1.0f) lo = mid; else hi = mid;
            }
            if (lane == 0) tau[node] = 0.5f * (lo + hi);
        }
    }
}

// ---- K5: emit out[e] = relu(score - smax[dst] - tau[dst]) in original order ----
__global__ void k_out(const float* __restrict__ logits, const int* __restrict__ dst,
                      const unsigned* __restrict__ smax_key, const float* __restrict__ tau,
                      float* __restrict__ out, int E) {
    int e = blockIdx.x * blockDim.x + threadIdx.x;
    if (e >= E) return;
    __builtin_prefetch(logits + e + 8192, 0, 3);
    __builtin_prefetch(dst + e + 8192, 0, 3);
    int d = dst[e];
    float v = logits[e] - key2f(smax_key[d]) - tau[d];
    out[e] = (v > 0.0f) ? v : 0.0f;
}

extern "C" void kernel_launch(void* const* d_in, const int* in_sizes, int n_in,
                              void* d_out, int out_size, void* d_ws, size_t ws_size,
                              hipStream_t stream) {
    const float* logits = (const float*)d_in[0];
    const int*   dst    = (const int*)d_in[1];
    const int*   dnum   = (const int*)d_in[2];   // device scalar: num_nodes
    float* out = (float*)d_out;
    const int E = in_sizes[0];

    // carve workspace: [gathered: E floats][smax_key|deg|cum|cnt|tau: per_node each]
    size_t ws_elems = ws_size / 4;
    size_t per_node = (ws_elems > (size_t)E) ? (ws_elems - (size_t)E) / 5 : 1;
    float*    gathered = (float*)d_ws;
    unsigned* smax_key = (unsigned*)(gathered + E);
    int*      deg      = (int*)(smax_key + per_node);
    int*      cum      = (int*)(deg + per_node);
    int*      cnt      = (int*)(cum + per_node);
    float*    tau      = (float*)(cnt + per_node);

    const int TB = 256;
    const int egrid = (E + TB - 1) / TB;

    k_init<<<2048, TB, 0, stream>>>(smax_key, deg, cnt, dnum);
    k_max_deg<<<egrid, TB, 0, stream>>>(logits, dst, smax_key, deg, E);
    k_scan<<<1, 1024, 0, stream>>>(deg, cum, dnum);
    k_scatter<<<egrid, TB, 0, stream>>>(logits, dst, smax_key, cum, cnt, gathered, E);
    k_tau<<<8192, 32, 0, stream>>>(gathered, cum, deg, tau, dnum);
    k_out<<<egrid, TB, 0, stream>>>(logits, dst, smax_key, tau, out, E);
}